// Cluster_loss_28733331210727
// MI455X (gfx1250) — compile-verified
//
#include <hip/hip_runtime.h>
#include <math.h>

typedef __attribute__((ext_vector_type(2))) float v2f;
typedef __attribute__((ext_vector_type(8))) float v8f;

#define NIMG 8
#define CCH  32
#define KC   16
#define HWP  262144          // 512*512
#define DELTA_V 0.2f
#define DELTA_D 0.2f

// ---------------------------------------------------------------------------
// Kernel 0: zero the workspace accumulators (sums, counts, distsum)
// ---------------------------------------------------------------------------
__global__ void k0_zero(float* __restrict__ ws, int nElem) {
    int i = blockIdx.x * 256 + threadIdx.x;
    if (i < nElem) ws[i] = 0.0f;
}

// ---------------------------------------------------------------------------
// Kernel 1: cluster sums + counts via WMMA F32 16x16x4 (one-hot GEMM).
// Grid: (256 pixel-blocks, NIMG). Block: 256 threads = 8 waves.
// Each wave streams 128 pixels in groups of 4:
//   A (16x4)  = features[c_base + 0..15][p..p+3]
//   B (4x16)  = onehot(gt[p..p+3], cluster 0..15)
//   D (16x16) = per-(channel, cluster) partial sums, chained accumulation.
// Two accumulators cover the 32 channels.
// ---------------------------------------------------------------------------
__global__ __launch_bounds__(256) void k1_sums(const float* __restrict__ f,
                                               const int*   __restrict__ gt,
                                               float* __restrict__ sums,
                                               float* __restrict__ counts) {
    __shared__ float lds_sums[CCH][KC];
    __shared__ float lds_cnt[KC];

    const int tid  = threadIdx.x;
    const int n    = blockIdx.y;
    const int wave = tid >> 5;
    const int lane = tid & 31;
    const int half = lane >> 4;   // 0: lanes 0-15, 1: lanes 16-31
    const int mod  = lane & 15;   // channel row (A) / cluster col (B,D)

    for (int i = tid; i < CCH * KC; i += 256) ((float*)lds_sums)[i] = 0.0f;
    if (tid < KC) lds_cnt[tid] = 0.0f;
    __syncthreads();

    const float* fb  = f  + (size_t)n * CCH * HWP;
    const int*   gb  = gt + (size_t)n * HWP;
    const int pWave  = blockIdx.x * 1024 + wave * 128;

    const float* fa0 = fb + (size_t)mod * HWP;          // channels 0..15
    const float* fa1 = fb + (size_t)(mod + 16) * HWP;   // channels 16..31

    v8f acc0 = {};   // channels 0..15  x clusters 0..15
    v8f acc1 = {};   // channels 16..31 x clusters 0..15
    float cnt = 0.0f;

    for (int g = 0; g < 128; g += 4) {
        const int p = pWave + g + 2 * half;   // even -> 8B aligned v2f loads
        const int g0 = gb[p];
        const int g1 = gb[p + 1];

        v2f b;  // one-hot columns for this lane's cluster `mod`
        b.x = (g0 == mod) ? 1.0f : 0.0f;
        b.y = (g1 == mod) ? 1.0f : 0.0f;
        cnt += b.x + b.y;

        v2f a0 = *(const v2f*)(fa0 + p);
        v2f a1 = *(const v2f*)(fa1 + p);

        acc0 = __builtin_amdgcn_wmma_f32_16x16x4_f32(false, a0, false, b,
                                                     (short)0, acc0, false, false);
        acc1 = __builtin_amdgcn_wmma_f32_16x16x4_f32(false, a1, false, b,
                                                     (short)0, acc1, false, false);
    }

    // D layout: lane holds D[v + 8*half][mod] in VGPR v
#pragma unroll
    for (int v = 0; v < 8; ++v) {
        atomicAdd(&lds_sums[v + 8 * half][mod],      acc0[v]);
        atomicAdd(&lds_sums[16 + v + 8 * half][mod], acc1[v]);
    }
    atomicAdd(&lds_cnt[mod], cnt);
    __syncthreads();

    for (int i = tid; i < CCH * KC; i += 256) {
        const int c = i / KC, k = i % KC;
        unsafeAtomicAdd(&sums[((size_t)n * CCH + c) * KC + k], lds_sums[c][k]);
    }
    if (tid < KC) unsafeAtomicAdd(&counts[n * KC + tid], lds_cnt[tid]);
}

// ---------------------------------------------------------------------------
// Kernel 2: means = sums / counts   (written straight into d_out+4)
// ---------------------------------------------------------------------------
__global__ void k2_means(const float* __restrict__ sums,
                         const float* __restrict__ counts,
                         float* __restrict__ means) {
    const int i = blockIdx.x * 256 + threadIdx.x;
    if (i >= NIMG * CCH * KC) return;
    const int k = i % KC;
    const int n = i / (CCH * KC);
    means[i] = sums[i] / counts[n * KC + k];
}

// ---------------------------------------------------------------------------
// Kernel 3: per-pixel distance to its cluster mean, reduced per (n,k).
// Grid: (HWP/256, NIMG). One thread = one pixel; 32 coalesced channel loads.
// ---------------------------------------------------------------------------
__global__ __launch_bounds__(256) void k3_dist(const float* __restrict__ f,
                                               const int*   __restrict__ gt,
                                               const float* __restrict__ means,
                                               float* __restrict__ distsum) {
    __shared__ float mu[KC][CCH + 1];   // +1 pad to dodge bank conflicts
    __shared__ float lds_d[KC];

    const int tid = threadIdx.x;
    const int n   = blockIdx.y;

    for (int i = tid; i < KC * CCH; i += 256) {
        const int c = i / KC, k = i % KC;
        mu[k][c] = means[((size_t)n * CCH + c) * KC + k];
    }
    if (tid < KC) lds_d[tid] = 0.0f;
    __syncthreads();

    const size_t p  = (size_t)blockIdx.x * 256 + tid;
    const float* fb = f + (size_t)n * CCH * HWP;
    const int    k  = gt[(size_t)n * HWP + p];

    float s = 0.0f;
#pragma unroll
    for (int c = 0; c < CCH; ++c) {
        const float d = fb[(size_t)c * HWP + p] - mu[k][c];
        s += d * d;
    }
    atomicAdd(&lds_d[k], sqrtf(s));
    __syncthreads();

    if (tid < KC) unsafeAtomicAdd(&distsum[n * KC + tid], lds_d[tid]);
}

// ---------------------------------------------------------------------------
// Kernel 4: finalize (single block, 128 threads = one per (n,k)).
// ---------------------------------------------------------------------------
__global__ __launch_bounds__(128) void k4_final(const float* __restrict__ counts,
                                                const float* __restrict__ distsum,
                                                const float* __restrict__ means,
                                                float* __restrict__ out) {
    __shared__ float varsum[NIMG];
    __shared__ float invsum[NIMG];
    __shared__ float hingeS;
    __shared__ float normS;

    const int t = threadIdx.x;
    if (t < NIMG) { varsum[t] = 0.0f; invsum[t] = 0.0f; }
    if (t == 0)   { hingeS = 0.0f; normS = 0.0f; }
    __syncthreads();

    const int n = t / KC, k = t % KC;

    float m[CCH];
#pragma unroll
    for (int c = 0; c < CCH; ++c) m[c] = means[((size_t)n * CCH + c) * KC + k];

    float nrm = 0.0f;
#pragma unroll
    for (int c = 0; c < CCH; ++c) nrm += m[c] * m[c];
    nrm = sqrtf(nrm);

    float pdsum = 0.0f;            // sum_j ||mu_k - mu_j|| (diag term is 0)
    for (int j = 0; j < KC; ++j) {
        float s = 0.0f;
#pragma unroll
        for (int c = 0; c < CCH; ++c) {
            const float d = m[c] - means[((size_t)n * CCH + c) * KC + j];
            s += d * d;
        }
        pdsum += sqrtf(s);
    }

    const float cntk   = counts[n * KC + k];
    const float meand  = distsum[n * KC + k] / cntk;
    const float var_nk = fmaxf(meand - DELTA_V, 0.0f) / cntk;

    atomicAdd(&varsum[n], var_nk);
    atomicAdd(&invsum[n], 1.0f / cntk);
    atomicAdd(&hingeS, fmaxf(2.0f * DELTA_D - pdsum / (float)(KC - 1), 0.0f));
    atomicAdd(&normS, nrm);
    __syncthreads();

    if (t == 0) {
        float vloss = 0.0f;
        for (int nn = 0; nn < NIMG; ++nn) vloss += varsum[nn] / invsum[nn];
        vloss /= (float)NIMG;
        const float dloss = hingeS / (float)(NIMG * KC);
        const float nloss = normS  / (float)(NIMG * KC);
        out[0] = vloss + dloss + 0.001f * nloss;
        out[1] = vloss;
        out[2] = dloss;
        out[3] = nloss;
    }
}

// ---------------------------------------------------------------------------
extern "C" void kernel_launch(void* const* d_in, const int* in_sizes, int n_in,
                              void* d_out, int out_size, void* d_ws, size_t ws_size,
                              hipStream_t stream) {
    (void)in_sizes; (void)n_in; (void)out_size; (void)ws_size;

    const float* f  = (const float*)d_in[0];   // [8,32,512,512] fp32
    const int*   gt = (const int*)d_in[1];     // [8,512,512] int32
    float* out = (float*)d_out;                // [4 scalars][means 8*32*16]

    float* sums   = (float*)d_ws;              // 4096 floats
    float* counts = sums + NIMG * CCH * KC;    // 128 floats
    float* dsum   = counts + NIMG * KC;        // 128 floats
    float* means  = out + 4;                   // means live in d_out directly

    const int nz = NIMG * CCH * KC + 2 * NIMG * KC;
    k0_zero<<<(nz + 255) / 256, 256, 0, stream>>>(sums, nz);

    k1_sums<<<dim3(HWP / 1024, NIMG), 256, 0, stream>>>(f, gt, sums, counts);
    k2_means<<<(NIMG * CCH * KC + 255) / 256, 256, 0, stream>>>(sums, counts, means);
    k3_dist<<<dim3(HWP / 256, NIMG), 256, 0, stream>>>(f, gt, means, dsum);
    k4_final<<<1, 128, 0, stream>>>(counts, dsum, means, out);
}